// YOLOLoss_5669356833741
// MI455X (gfx1250) — compile-verified
//
#include <hip/hip_runtime.h>
#include <math.h>

#define B_   8
#define G_   50
#define A0_  6400
#define A1_  1600
#define A2_  400
#define A_   8400
#define CCH  85
#define NC   80

typedef float v2f __attribute__((ext_vector_type(2)));
typedef float v8f __attribute__((ext_vector_type(8)));

// ---------------- helpers ----------------

__device__ __forceinline__ float feat_at(const float* __restrict__ f0,
                                         const float* __restrict__ f1,
                                         const float* __restrict__ f2,
                                         int b, int a, int c) {
  if (a < A0_) {
    int y = a / 80, x = a - y * 80;
    return f0[((b * CCH + c) * 80 + y) * 80 + x];
  } else if (a < A0_ + A1_) {
    int i = a - A0_;
    int y = i / 40, x = i - y * 40;
    return f1[((b * CCH + c) * 40 + y) * 40 + x];
  } else {
    int i = a - (A0_ + A1_);
    int y = i / 20, x = i - y * 20;
    return f2[((b * CCH + c) * 20 + y) * 20 + x];
  }
}

__device__ __forceinline__ void anchor_geom(int a, float& st, float& gx, float& gy) {
  if (a < A0_)            { st = 8.f;  int y = a / 80;            gx = (float)(a - y * 80); gy = (float)y; }
  else if (a < A0_ + A1_) { st = 16.f; int i = a - A0_;           int y = i / 40; gx = (float)(i - y * 40); gy = (float)y; }
  else                    { st = 32.f; int i = a - (A0_ + A1_);   int y = i / 20; gx = (float)(i - y * 20); gy = (float)y; }
}

__device__ __forceinline__ float bce_logits(float x, float t) {
  return fmaxf(x, 0.f) - x * t + log1pf(expf(-fabsf(x)));
}

// ---------------- kernel 1: decode bboxes ----------------

__global__ void decode_kernel(const float* __restrict__ f0, const float* __restrict__ f1,
                              const float* __restrict__ f2, float* __restrict__ bbox) {
  int idx = blockIdx.x * blockDim.x + threadIdx.x;
  if (idx >= B_ * A_) return;
  int b = idx / A_, a = idx - b * A_;
  float st, gx, gy; anchor_geom(a, st, gx, gy);
  float t0 = feat_at(f0, f1, f2, b, a, 0);
  float t1 = feat_at(f0, f1, f2, b, a, 1);
  float t2 = feat_at(f0, f1, f2, b, a, 2);
  float t3 = feat_at(f0, f1, f2, b, a, 3);
  float4 r;
  r.x = (t0 + gx) * st;
  r.y = (t1 + gy) * st;
  r.z = expf(t2) * st;
  r.w = expf(t3) * st;
  reinterpret_cast<float4*>(bbox)[idx] = r;
}

// ---------------- kernel 2: sum_neg via f32 WMMA row-sum ----------------
// sum_neg[b][a] = -sum_c max(log1p(-s[a,c]), -100),  s = sqrt(sig(cls)*sig(obj))
// One wave handles a tile of 16 anchors. A-matrix 16x4 f32 per step, B = ones,
// accumulate over 20 fully-unrolled steps (80 classes): loads get hoisted and
// the sigmoid/log VALU chain co-executes with the WMMA (XDL) pipe.
// D row-sums sit in lanes 0 (M=0..7) and 16 (M=8..15) per the 16x16 f32 layout.

__global__ void sumneg_wmma_kernel(const float* __restrict__ f0, const float* __restrict__ f1,
                                   const float* __restrict__ f2, float* __restrict__ sumneg) {
  int gtid = blockIdx.x * blockDim.x + threadIdx.x;
  int wave = gtid >> 5;
  int lane = gtid & 31;
  int b    = wave / (A_ / 16);
  int tile = wave - b * (A_ / 16);
  int base = tile * 16;
  int m    = lane & 15;
  int half = lane >> 4;       // 0: K=0,1  1: K=2,3
  int anchor = base + m;

  float xo = feat_at(f0, f1, f2, b, anchor, 4);
  float so = 1.f / (1.f + expf(-xo));

  v8f acc = {0.f, 0.f, 0.f, 0.f, 0.f, 0.f, 0.f, 0.f};
  v2f ones; ones.x = 1.f; ones.y = 1.f;

#pragma unroll
  for (int kk = 0; kk < 20; ++kk) {
    int c0 = kk * 4 + half * 2;
    float x0 = feat_at(f0, f1, f2, b, anchor, 5 + c0);
    float x1 = feat_at(f0, f1, f2, b, anchor, 5 + c0 + 1);
    float s0 = sqrtf((1.f / (1.f + expf(-x0))) * so);
    float s1 = sqrtf((1.f / (1.f + expf(-x1))) * so);
    v2f av;
    av.x = fmaxf(log1pf(-s0), -100.f);
    av.y = fmaxf(log1pf(-s1), -100.f);
    acc = __builtin_amdgcn_wmma_f32_16x16x4_f32(false, av, false, ones,
                                                (short)0, acc, false, false);
  }

  if (m == 0) {
    float* dst = sumneg + b * A_ + base + half * 8;
    dst[0] = -acc[0]; dst[1] = -acc[1]; dst[2] = -acc[2]; dst[3] = -acc[3];
    dst[4] = -acc[4]; dst[5] = -acc[5]; dst[6] = -acc[6]; dst[7] = -acc[7];
  }
}

// ---------------- kernel 3: fg0 mask ----------------

__global__ void fg0_kernel(const float* __restrict__ labels, float* __restrict__ fg0) {
  int idx = blockIdx.x * blockDim.x + threadIdx.x;
  if (idx >= B_ * A_) return;
  int b = idx / A_, a = idx - b * A_;
  float st, gx, gy; anchor_geom(a, st, gx, gy);
  float xc = (gx + 0.5f) * st, yc = (gy + 0.5f) * st, r = 2.5f * st;
  bool anyb = false, anyc = false;
  for (int g = 0; g < G_; ++g) {
    const float* gt = labels + (b * G_ + g) * 5;
    float cx = gt[0], cy = gt[1], w = gt[2], h = gt[3];
    bool ib = (xc - (cx - 0.5f * w) > 0.f) && ((cx + 0.5f * w) - xc > 0.f) &&
              (yc - (cy - 0.5f * h) > 0.f) && ((cy + 0.5f * h) - yc > 0.f);
    bool ic = (xc - (cx - r) > 0.f) && ((cx + r) - xc > 0.f) &&
              (yc - (cy - r) > 0.f) && ((cy + r) - yc > 0.f);
    anyb |= ib; anyc |= ic;
  }
  fg0[idx] = (anyb || anyc) ? 1.f : 0.f;
}

// ---------------- kernel 4: pairwise IoU + SimOTA cost ----------------

__global__ void cost_kernel(const float* __restrict__ f0, const float* __restrict__ f1,
                            const float* __restrict__ f2, const float* __restrict__ labels,
                            const float* __restrict__ bbox, const float* __restrict__ sumneg,
                            const float* __restrict__ fg0,
                            float* __restrict__ ious, float* __restrict__ cost) {
  int idx = blockIdx.x * blockDim.x + threadIdx.x;
  if (idx >= B_ * G_ * A_) return;
  int a = idx % A_;
  int t = idx / A_;
  int g = t % G_;
  int b = t / G_;

  const float* gt = labels + (b * G_ + g) * 5;
  float cx = gt[0], cy = gt[1], gw = gt[2], gh = gt[3];
  int cg = (int)gt[4];

  float4 p = reinterpret_cast<const float4*>(bbox)[b * A_ + a];

  // pairwise IoU (no epsilon, matching reference)
  float tlx = fmaxf(cx - gw * 0.5f, p.x - p.z * 0.5f);
  float tly = fmaxf(cy - gh * 0.5f, p.y - p.w * 0.5f);
  float brx = fminf(cx + gw * 0.5f, p.x + p.z * 0.5f);
  float bry = fminf(cy + gh * 0.5f, p.y + p.w * 0.5f);
  float en  = (tlx < brx && tly < bry) ? 1.f : 0.f;
  float ai  = (brx - tlx) * (bry - tly) * en;
  float iou = ai / (gw * gh + p.z * p.w - ai);
  ious[idx] = iou;
  float iou_cost = -logf(iou + 1e-8f);

  // cls cost for this gt's class
  float xo = feat_at(f0, f1, f2, b, a, 4);
  float xc_ = feat_at(f0, f1, f2, b, a, 5 + cg);
  float so = 1.f / (1.f + expf(-xo));
  float sc = 1.f / (1.f + expf(-xc_));
  float s  = sqrtf(sc * so);
  float lns = fmaxf(logf(s), -100.f);
  float l1m = fmaxf(log1pf(-s), -100.f);
  float cls_cost = sumneg[b * A_ + a] + l1m - lns;

  // geometric "both" penalty
  float st, gx, gy; anchor_geom(a, st, gx, gy);
  float ax = (gx + 0.5f) * st, ay = (gy + 0.5f) * st, r = 2.5f * st;
  bool ib = (ax - (cx - 0.5f * gw) > 0.f) && ((cx + 0.5f * gw) - ax > 0.f) &&
            (ay - (cy - 0.5f * gh) > 0.f) && ((cy + 0.5f * gh) - ay > 0.f);
  bool ic = (ax - (cx - r) > 0.f) && ((cx + r) - ax > 0.f) &&
            (ay - (cy - r) > 0.f) && ((cy + r) - ay > 0.f);
  bool both = ib && ic;

  float cst = cls_cost + 3.f * iou_cost + (both ? 0.f : 100000.f) +
              (fg0[b * A_ + a] > 0.f ? 0.f : 1000000000.f);
  cost[idx] = cst;
}

// ---------------- kernel 5: per-gt dynamic-k selection ----------------
// Register-resident sorted-10 lists via fully unrolled bubble-insert
// (constant indices -> stays in VGPRs, no scratch spill).

__device__ __forceinline__ void keep_max10(float* t, float v) {
#pragma unroll
  for (int i = 0; i < 10; ++i) {
    if (v > t[i]) { float tmp = t[i]; t[i] = v; v = tmp; }
  }
}
__device__ __forceinline__ void keep_min10(float* t, float v) {
#pragma unroll
  for (int i = 0; i < 10; ++i) {
    if (v < t[i]) { float tmp = t[i]; t[i] = v; v = tmp; }
  }
}

__global__ void select_kernel(const float* __restrict__ ious, const float* __restrict__ cost,
                              const float* __restrict__ fg0, float* __restrict__ kth) {
  int bg = blockIdx.x;            // b*G_ + g
  int b  = bg / G_;
  const float* irow = ious + (size_t)bg * A_;
  const float* crow = cost + (size_t)bg * A_;
  const float* fgr  = fg0 + b * A_;

  float ti[10], tc[10];
#pragma unroll
  for (int i = 0; i < 10; ++i) { ti[i] = -1e30f; tc[i] = 3.4e38f; }

  for (int a = threadIdx.x; a < A_; a += blockDim.x) {
    int an = a + blockDim.x;
    if (an < A_) {                      // hide L2 latency of the next stride
      __builtin_prefetch(&irow[an], 0, 3);
      __builtin_prefetch(&crow[an], 0, 3);
    }
    keep_max10(ti, irow[a] * fgr[a]);
    keep_min10(tc, crow[a]);
  }

  __shared__ float si[128 * 10];
  __shared__ float sc[128 * 10];
#pragma unroll
  for (int i = 0; i < 10; ++i) {
    si[threadIdx.x * 10 + i] = ti[i];
    sc[threadIdx.x * 10 + i] = tc[i];
  }
  __syncthreads();

  if (threadIdx.x == 0) {
    float bi[10], bc[10];
#pragma unroll
    for (int i = 0; i < 10; ++i) { bi[i] = -1e30f; bc[i] = 3.4e38f; }
    for (int t = 0; t < 128 * 10; ++t) {
      keep_max10(bi, si[t]);
      keep_min10(bc, sc[t]);
    }
    float s = 0.f;
#pragma unroll
    for (int i = 0; i < 10; ++i) s += bi[i];
    int dk = (int)s;                 // truncation, matching astype(int32)
    if (dk < 1) dk = 1;
    if (dk > 10) dk = 10;
    float kv = bc[0];
#pragma unroll
    for (int i = 0; i < 10; ++i) if (i == dk - 1) kv = bc[i];
    kth[bg] = kv;
  }
}

// ---------------- kernel 6: matching + conflict resolution ----------------

__global__ void match_kernel(const float* __restrict__ ious, const float* __restrict__ cost,
                             const float* __restrict__ kth,
                             float* __restrict__ fg, int* __restrict__ mg,
                             float* __restrict__ miou) {
  int idx = blockIdx.x * blockDim.x + threadIdx.x;
  if (idx >= B_ * A_) return;
  int b = idx / A_, a = idx - b * A_;

  int   cnt = 0, first = 0, bestg = 0;
  float msum = 0.f, bestc = 3.4e38f, besti = 0.f;
  bool  havef = false;

  for (int g = 0; g < G_; ++g) {
    int o = (b * G_ + g) * A_ + a;
    float c = cost[o];
    float i = ious[o];
    if (c <= kth[b * G_ + g]) {
      cnt++; msum += i;
      if (!havef) { first = g; havef = true; }
    }
    if (c < bestc) { bestc = c; bestg = g; besti = i; }
  }

  if (cnt > 1)      { fg[idx] = 1.f; mg[idx] = bestg; miou[idx] = besti; }
  else if (cnt == 1){ fg[idx] = 1.f; mg[idx] = first; miou[idx] = msum; }
  else              { fg[idx] = 0.f; mg[idx] = 0;     miou[idx] = 0.f;  }
}

// ---------------- kernel 7: loss partials ----------------

__global__ void loss_kernel(const float* __restrict__ f0, const float* __restrict__ f1,
                            const float* __restrict__ f2, const float* __restrict__ labels,
                            const float* __restrict__ bbox, const float* __restrict__ fg,
                            const int* __restrict__ mg, const float* __restrict__ miou,
                            float* __restrict__ partials, int nb) {
  int idx = blockIdx.x * blockDim.x + threadIdx.x;
  float tot = 0.f, nf = 0.f;

  if (idx < B_ * A_) {
    int b = idx / A_, a = idx - b * A_;
    float fgf = fg[idx];

    float xo = feat_at(f0, f1, f2, b, a, 4);
    tot += bce_logits(xo, fgf);                    // loss_obj

    if (fgf > 0.f) {
      nf = 1.f;
      float4 p = reinterpret_cast<const float4*>(bbox)[idx];
      const float* gt = labels + (b * G_ + mg[idx]) * 5;
      float cx = gt[0], cy = gt[1], gw = gt[2], gh = gt[3];

      float tlx = fmaxf(p.x - p.z * 0.5f, cx - gw * 0.5f);
      float tly = fmaxf(p.y - p.w * 0.5f, cy - gh * 0.5f);
      float brx = fminf(p.x + p.z * 0.5f, cx + gw * 0.5f);
      float bry = fminf(p.y + p.w * 0.5f, cy + gh * 0.5f);
      float en  = (tlx < brx && tly < bry) ? 1.f : 0.f;
      float ai  = (brx - tlx) * (bry - tly) * en;
      float iou = ai / (p.z * p.w + gw * gh - ai + 1e-16f);
      tot += 5.f * (1.f - iou * iou);              // 5 * loss_iou

      int   mc = (int)gt[4];
      float mi = miou[idx];
      float cs = 0.f;
      for (int c = 0; c < NC; ++c) {
        float x = feat_at(f0, f1, f2, b, a, 5 + c);
        float t = (c == mc) ? mi : 0.f;
        cs += bce_logits(x, t);
      }
      tot += cs;                                   // loss_cls
    }
  }

  __shared__ float s1[256];
  __shared__ float s2[256];
  s1[threadIdx.x] = tot; s2[threadIdx.x] = nf;
  __syncthreads();
  for (int st = 128; st > 0; st >>= 1) {
    if (threadIdx.x < st) {
      s1[threadIdx.x] += s1[threadIdx.x + st];
      s2[threadIdx.x] += s2[threadIdx.x + st];
    }
    __syncthreads();
  }
  if (threadIdx.x == 0) {
    partials[blockIdx.x]      = s1[0];
    partials[nb + blockIdx.x] = s2[0];
  }
}

// ---------------- kernel 8: finalize ----------------

__global__ void finalize_kernel(const float* __restrict__ partials, int nb,
                                float* __restrict__ out) {
  __shared__ float s1[256];
  __shared__ float s2[256];
  float t = 0.f, n = 0.f;
  for (int i = threadIdx.x; i < nb; i += 256) {
    t += partials[i];
    n += partials[nb + i];
  }
  s1[threadIdx.x] = t; s2[threadIdx.x] = n;
  __syncthreads();
  for (int st = 128; st > 0; st >>= 1) {
    if (threadIdx.x < st) {
      s1[threadIdx.x] += s1[threadIdx.x + st];
      s2[threadIdx.x] += s2[threadIdx.x + st];
    }
    __syncthreads();
  }
  if (threadIdx.x == 0) out[0] = s1[0] / fmaxf(s2[0], 1.f);
}

// ---------------- host launch ----------------

extern "C" void kernel_launch(void* const* d_in, const int* in_sizes, int n_in,
                              void* d_out, int out_size, void* d_ws, size_t ws_size,
                              hipStream_t stream) {
  const float* f0     = (const float*)d_in[0];
  const float* f1     = (const float*)d_in[1];
  const float* f2     = (const float*)d_in[2];
  const float* labels = (const float*)d_in[3];

  char* w = (char*)d_ws;
  size_t off = 0;
  auto take = [&](size_t bytes) -> void* {
    void* p = w + off;
    off += (bytes + 255) & ~(size_t)255;
    return p;
  };

  float* bbox    = (float*)take((size_t)B_ * A_ * 4 * sizeof(float));
  float* sumneg  = (float*)take((size_t)B_ * A_ * sizeof(float));
  float* fg0     = (float*)take((size_t)B_ * A_ * sizeof(float));
  float* ious    = (float*)take((size_t)B_ * G_ * A_ * sizeof(float));
  float* cost    = (float*)take((size_t)B_ * G_ * A_ * sizeof(float));
  float* kth     = (float*)take((size_t)B_ * G_ * sizeof(float));
  float* fg      = (float*)take((size_t)B_ * A_ * sizeof(float));
  int*   mg      = (int*)  take((size_t)B_ * A_ * sizeof(int));
  float* miou    = (float*)take((size_t)B_ * A_ * sizeof(float));

  const int NBA = (B_ * A_ + 255) / 256;                  // 263
  float* partials = (float*)take((size_t)2 * NBA * sizeof(float));

  decode_kernel<<<NBA, 256, 0, stream>>>(f0, f1, f2, bbox);

  // exactly B_*(A_/16) = 4200 waves; 8 waves/block -> 525 full blocks (EXEC all-1s)
  sumneg_wmma_kernel<<<(B_ * (A_ / 16) * 32) / 256, 256, 0, stream>>>(f0, f1, f2, sumneg);

  fg0_kernel<<<NBA, 256, 0, stream>>>(labels, fg0);

  cost_kernel<<<(B_ * G_ * A_ + 255) / 256, 256, 0, stream>>>(f0, f1, f2, labels, bbox,
                                                              sumneg, fg0, ious, cost);

  select_kernel<<<B_ * G_, 128, 0, stream>>>(ious, cost, fg0, kth);

  match_kernel<<<NBA, 256, 0, stream>>>(ious, cost, kth, fg, mg, miou);

  loss_kernel<<<NBA, 256, 0, stream>>>(f0, f1, f2, labels, bbox, fg, mg, miou,
                                       partials, NBA);

  finalize_kernel<<<1, 256, 0, stream>>>(partials, NBA, (float*)d_out);
}